// SPDLogMap_8615704396364
// MI455X (gfx1250) — compile-verified
//
#include <hip/hip_runtime.h>

typedef __attribute__((ext_vector_type(2))) float v2f;
typedef __attribute__((ext_vector_type(8))) float v8f;

#define NN 64
#define LDSP 65          // padded row stride (in float2): dword bank = 2(i+j) & 63
#define NSWEEPS 10
#define EPSV 1e-5f

__device__ __forceinline__ void jacobi_pair(int k, int r, int& p, int& q) {
    // round-robin tournament pairing: 63 rounds x 32 pairs covers all C(64,2) pairs
    if (k == 0) { p = NN - 1; q = r % (NN - 1); }
    else        { p = (r + k) % (NN - 1); q = (r + (NN - 1) - k) % (NN - 1); }
    if (p > q) { int t = p; p = q; q = t; }
}

__global__ __launch_bounds__(256)
void spd_logmap_kernel(const float* __restrict__ S, float* __restrict__ out)
{
    // W[i][j].x = A_ij (working matrix, later T = V*diag(logw))
    // W[i][j].y = V_ij (accumulated eigenvectors, columns)
    __shared__ float2 W[NN * LDSP];
    __shared__ float  lw[NN];

    const int tid = threadIdx.x;
    const long long b = blockIdx.x;
    const float* __restrict__ Sb = S + b * (long long)(NN * NN);
    float* __restrict__ Ob = out + b * (long long)(NN * NN);

    // ---- load S (coalesced b128) into padded LDS; V = I ----
    for (int t = tid; t < NN * NN / 4; t += 256) {
        float4 v = ((const float4*)Sb)[t];
        int idx = t * 4;
        int i = idx >> 6, j = idx & 63;
        W[i * LDSP + j + 0] = make_float2(v.x, (i == j + 0) ? 1.0f : 0.0f);
        W[i * LDSP + j + 1] = make_float2(v.y, (i == j + 1) ? 1.0f : 0.0f);
        W[i * LDSP + j + 2] = make_float2(v.z, (i == j + 2) ? 1.0f : 0.0f);
        W[i * LDSP + j + 3] = make_float2(v.w, (i == j + 3) ? 1.0f : 0.0f);
    }
    __syncthreads();

    // ---- parallel cyclic Jacobi: A' = J^T A J over 32 disjoint pairs/round ----
    const int k = tid >> 3;                       // 8 threads per pair
    for (int sweep = 0; sweep < NSWEEPS; ++sweep) {
        for (int r = 0; r < NN - 1; ++r) {
            int p, q; jacobi_pair(k, r, p, q);

            // all 8 threads of a pair compute (c,s) redundantly (LDS broadcast reads)
            float app = W[p * LDSP + p].x;
            float aqq = W[q * LDSP + q].x;
            float apq = W[p * LDSP + q].x;
            float c = 1.0f, s = 0.0f;
            if (fabsf(apq) > 1e-24f) {
                float tau = (aqq - app) / (2.0f * apq);
                float t = copysignf(1.0f, tau) / (fabsf(tau) + sqrtf(1.0f + tau * tau));
                c = rsqrtf(1.0f + t * t);
                s = t * c;
            }
            __syncthreads();                      // rotation inputs read before writes

            // column update: [A|V] <- [A|V]*J  (b64 per element pair, 8 rows/thread)
            #pragma unroll
            for (int jj = 0; jj < 8; ++jj) {
                int i = (tid & 7) + (jj << 3);
                float2 wp = W[i * LDSP + p];
                float2 wq = W[i * LDSP + q];
                W[i * LDSP + p] = make_float2(c * wp.x - s * wq.x, c * wp.y - s * wq.y);
                W[i * LDSP + q] = make_float2(s * wp.x + c * wq.x, s * wp.y + c * wq.y);
            }
            __syncthreads();

            // row update: A <- J^T * A  (scalar .x only)
            #pragma unroll
            for (int jj = 0; jj < 8; ++jj) {
                int j = (tid & 7) + (jj << 3);
                float x = W[p * LDSP + j].x;
                float y = W[q * LDSP + j].x;
                W[p * LDSP + j].x = c * x - s * y;
                W[q * LDSP + j].x = s * x + c * y;
            }
            __syncthreads();
        }
    }

    // ---- log-eigenvalues, then T = V * diag(log w) stored into .x ----
    if (tid < NN) {
        float w = W[tid * LDSP + tid].x;
        lw[tid] = logf(fmaxf(w, EPSV));
    }
    __syncthreads();
    for (int t = tid; t < NN * NN; t += 256) {
        int i = t >> 6, j = t & 63;
        float2 wv = W[i * LDSP + j];
        W[i * LDSP + j].x = wv.y * lw[j];
    }
    __syncthreads();

    // ---- OUT = T * V^T via V_WMMA_F32_16X16X4_F32 ----
    // 16 output tiles (4x4 grid of 16x16); 8 waves x 2 tiles; 16 K-steps of 4.
    const int lane = tid & 31;
    const int wave = tid >> 5;
    const int half = lane >> 4;    // lane group: 0 -> lanes 0-15, 1 -> lanes 16-31
    const int mn   = lane & 15;

    for (int tt = 0; tt < 2; ++tt) {
        int tile = wave * 2 + tt;
        int mi = tile >> 2;        // row tile
        int nj = tile & 3;         // col tile
        v8f acc = {};
        #pragma unroll
        for (int kk = 0; kk < 16; ++kk) {
            int K0 = kk * 4 + half * 2;
            // A frag (16x4, T in .x): lanes0-15 K={0,1}, lanes16-31 K={2,3}
            v2f af, bf;
            af.x = W[(mi * 16 + mn) * LDSP + K0 + 0].x;
            af.y = W[(mi * 16 + mn) * LDSP + K0 + 1].x;
            // B frag (4x16) = V^T (V in .y): row K striped over 16 lanes; K = vgpr + 2*half
            bf.x = W[(nj * 16 + mn) * LDSP + K0 + 0].y;
            bf.y = W[(nj * 16 + mn) * LDSP + K0 + 1].y;
            acc = __builtin_amdgcn_wmma_f32_16x16x4_f32(
                      false, af, false, bf, (short)0, acc, false, false);
        }
        // D layout: VGPR r -> M = r + 8*half, N = mn ; 16-lane contiguous stores
        #pragma unroll
        for (int rr = 0; rr < 8; ++rr) {
            int row = mi * 16 + rr + half * 8;
            Ob[row * 64 + nj * 16 + mn] = acc[rr];
        }
    }
}

extern "C" void kernel_launch(void* const* d_in, const int* in_sizes, int n_in,
                              void* d_out, int out_size, void* d_ws, size_t ws_size,
                              hipStream_t stream) {
    (void)n_in; (void)out_size; (void)d_ws; (void)ws_size;
    const float* S = (const float*)d_in[0];
    float* out = (float*)d_out;
    int B = in_sizes[0] / (NN * NN);   // 8192
    spd_logmap_kernel<<<dim3(B), dim3(256), 0, stream>>>(S, out);
}